// GNN_62491774157019
// MI455X (gfx1250) — compile-verified
//
#include <hip/hip_runtime.h>
#include <hip/hip_bf16.h>

typedef __attribute__((ext_vector_type(2))) float v2f;
typedef __attribute__((ext_vector_type(8))) float v8f;

// ---------------------------------------------------------------------------
// GEMM1: h1[N,16] = x[N,128] @ W1[128,16]   (f32 WMMA 16x16x4, K-loop of 32)
// One 16-row tile per wave; 4 waves per block.
// A-frag (16x4 f32): lanes 0-15 -> M=lane, K = k0,k0+1 ; lanes 16-31 -> K = k0+2,k0+3
// B-frag (4x16 f32): lanes 0-15 -> N=lane, K = k0,k0+1 ; lanes 16-31 -> K = k0+2,k0+3
// C/D (16x16 f32): VGPR r: lanes 0-15 -> M=r, lanes 16-31 -> M=r+8
// ---------------------------------------------------------------------------
__global__ void gemm1_wmma(const float* __restrict__ x,
                           const float* __restrict__ W1,
                           float* __restrict__ h1, int n_nodes) {
    const int lane = threadIdx.x & 31;
    const int wave = threadIdx.x >> 5;
    const int tile = blockIdx.x * (blockDim.x >> 5) + wave;
    const int tiles = (n_nodes + 15) >> 4;
    if (tile >= tiles) return;                 // wave-uniform exit

    const int half = lane >> 4;                // 0: lanes 0-15, 1: lanes 16-31
    const int n    = lane & 15;                // output column / A row index

    int row = tile * 16 + n;
    if (row >= n_nodes) row = n_nodes - 1;     // clamp: keep EXEC all-ones
    const float* __restrict__ xrow = x + (size_t)row * 128;

    v8f acc = {};
#pragma unroll
    for (int k0 = 0; k0 < 128; k0 += 4) {
        const int ka = k0 + 2 * half;
        v2f a, b;
        a.x = xrow[ka];
        a.y = xrow[ka + 1];
        b.x = W1[(size_t)ka * 16 + n];
        b.y = W1[(size_t)(ka + 1) * 16 + n];
        acc = __builtin_amdgcn_wmma_f32_16x16x4_f32(
            /*neg_a=*/false, a, /*neg_b=*/false, b,
            /*c_mod=*/(short)0, acc, /*reuse_a=*/false, /*reuse_b=*/false);
    }

#pragma unroll
    for (int r = 0; r < 8; ++r) {
        const int mo   = r + 8 * half;
        const int orow = tile * 16 + mo;
        if (orow < n_nodes) h1[(size_t)orow * 16 + n] = acc[r];
    }
}

// ---------------------------------------------------------------------------
// GEMM2 (fused): h2[N,16] = relu(agg1[N,16] + b1) @ W2[16,16]  (4 WMMAs)
// ---------------------------------------------------------------------------
__global__ void gemm2_wmma(const float* __restrict__ agg1,
                           const float* __restrict__ b1,
                           const float* __restrict__ W2,
                           float* __restrict__ h2, int n_nodes) {
    const int lane = threadIdx.x & 31;
    const int wave = threadIdx.x >> 5;
    const int tile = blockIdx.x * (blockDim.x >> 5) + wave;
    const int tiles = (n_nodes + 15) >> 4;
    if (tile >= tiles) return;

    const int half = lane >> 4;
    const int n    = lane & 15;

    int row = tile * 16 + n;
    if (row >= n_nodes) row = n_nodes - 1;
    const float* __restrict__ arow = agg1 + (size_t)row * 16;

    v8f acc = {};
#pragma unroll
    for (int k0 = 0; k0 < 16; k0 += 4) {
        const int ka = k0 + 2 * half;
        v2f a, b;
        float a0 = arow[ka]     + b1[ka];
        float a1 = arow[ka + 1] + b1[ka + 1];
        a.x = a0 > 0.0f ? a0 : 0.0f;           // fused bias + relu from layer 1
        a.y = a1 > 0.0f ? a1 : 0.0f;
        b.x = W2[(size_t)ka * 16 + n];
        b.y = W2[(size_t)(ka + 1) * 16 + n];
        acc = __builtin_amdgcn_wmma_f32_16x16x4_f32(
            false, a, false, b, (short)0, acc, false, false);
    }

#pragma unroll
    for (int r = 0; r < 8; ++r) {
        const int mo   = r + 8 * half;
        const int orow = tile * 16 + mo;
        if (orow < n_nodes) h2[(size_t)orow * 16 + n] = acc[r];
    }
}

// ---------------------------------------------------------------------------
// Edge scatter: agg[dst] += h[src] * w   — 16 lanes per edge so the 16 f32
// atomics per edge are a contiguous 64B row (coalesced global_atomic_add_f32,
// L2-resident since agg is 12.8 MB << 192 MB L2).
// ---------------------------------------------------------------------------
__global__ void edge_scatter(const float* __restrict__ h,
                             const int* __restrict__ src,
                             const int* __restrict__ dst,
                             const float* __restrict__ w,
                             float* __restrict__ agg, int n_edges) {
    const long long t = (long long)blockIdx.x * blockDim.x + threadIdx.x;
    const long long e = t >> 4;
    const int c = (int)(t & 15);
    if (e < n_edges) {
        const int s  = src[e];
        const int d  = dst[e];
        const float wt = w[e];
        const float v  = h[(size_t)s * 16 + c] * wt;
        atomicAdd(&agg[(size_t)d * 16 + c], v);
    }
}

// ---------------------------------------------------------------------------
// Final dense: out[n] = relu(agg2[n] + b2) . Wd + bd
// ---------------------------------------------------------------------------
__global__ void final_dense(const float* __restrict__ agg2,
                            const float* __restrict__ b2,
                            const float* __restrict__ Wd,
                            const float* __restrict__ bd,
                            float* __restrict__ out, int n_nodes) {
    const int i = blockIdx.x * blockDim.x + threadIdx.x;
    if (i >= n_nodes) return;
    const float* __restrict__ row = agg2 + (size_t)i * 16;
    float acc = bd[0];
#pragma unroll
    for (int c = 0; c < 16; ++c) {
        float hv = row[c] + b2[c];
        hv = hv > 0.0f ? hv : 0.0f;
        acc += hv * Wd[c];
    }
    out[i] = acc;
}

// ---------------------------------------------------------------------------
// Launch
// ---------------------------------------------------------------------------
extern "C" void kernel_launch(void* const* d_in, const int* in_sizes, int n_in,
                              void* d_out, int out_size, void* d_ws, size_t ws_size,
                              hipStream_t stream) {
    const float* x  = (const float*)d_in[0];
    const int*   es = (const int*)  d_in[1];
    const int*   ed = (const int*)  d_in[2];
    const float* ew = (const float*)d_in[3];
    const float* W1 = (const float*)d_in[4];
    const float* b1 = (const float*)d_in[5];
    const float* W2 = (const float*)d_in[6];
    const float* b2 = (const float*)d_in[7];
    const float* Wd = (const float*)d_in[8];
    const float* bd = (const float*)d_in[9];
    float* out = (float*)d_out;

    const int n_nodes = in_sizes[0] / 128;   // 200000
    const int n_edges = in_sizes[1];         // 6400000
    const size_t nh = (size_t)n_nodes * 16;

    float* h1   = (float*)d_ws;              // [N,16]
    float* agg1 = h1 + nh;                   // [N,16]
    float* h2   = agg1 + nh;                 // [N,16]
    float* agg2 = h2 + nh;                   // [N,16]

    hipMemsetAsync(agg1, 0, nh * sizeof(float), stream);
    hipMemsetAsync(agg2, 0, nh * sizeof(float), stream);

    const int tiles = (n_nodes + 15) / 16;            // 12500
    const int waves_per_block = 4;                    // 128 threads
    const int gblocks = (tiles + waves_per_block - 1) / waves_per_block;
    dim3 gblk(32 * waves_per_block);

    gemm1_wmma<<<gblocks, gblk, 0, stream>>>(x, W1, h1, n_nodes);

    const long long ethreads = (long long)n_edges * 16;
    const int eblocks = (int)((ethreads + 255) / 256);
    edge_scatter<<<eblocks, 256, 0, stream>>>(h1, es, ed, ew, agg1, n_edges);

    gemm2_wmma<<<gblocks, gblk, 0, stream>>>(agg1, b1, W2, h2, n_nodes);

    edge_scatter<<<eblocks, 256, 0, stream>>>(h2, es, ed, ew, agg2, n_edges);

    const int fblocks = (n_nodes + 255) / 256;
    final_dense<<<fblocks, 256, 0, stream>>>(agg2, b2, Wd, bd, out, n_nodes);
}